// SVDSECON_75058848465569
// MI455X (gfx1250) — compile-verified
//
#include <hip/hip_runtime.h>
#include <hip/hip_bf16.h>
#include <math.h>

// Problem constants (X: [64, 1024, 256] f32; m > n branch of reference)
#define BATCH 64
#define M 1024
#define N 256
#define MAX_ITER 8

typedef float v2f __attribute__((ext_vector_type(2)));
typedef float v8f __attribute__((ext_vector_type(8)));

// D = A(16x4 f32) * B(4x16 f32) + C(16x16 f32), wave32 WMMA.
__device__ __forceinline__ v8f wmma_f32_4(v2f a, v2f b, v8f c) {
    return __builtin_amdgcn_wmma_f32_16x16x4_f32(
        /*neg_a=*/false, a, /*neg_b=*/false, b,
        /*c_mod=*/(short)0, c, /*reuse_a=*/false, /*reuse_b=*/false);
}

// WMMA f32 16x16x4 fragment layout (ISA 7.12.2):
//   A 16x4 : lane<16 -> {K=k0,k0+1}, lane>=16 -> {K=k0+2,k0+3}, M = lane&15
//   B 4x16 : mirrored (N = lane&15)
//   C 16x16: vgpr r -> M = r + 8*(lane>>4), N = lane&15

// ---------------------------------------------------------------------------
// Kernel 1: A = X^T X per batch (256x256x1024 GEMM). Also Z = A (iteration 0:
// Z = A @ I). Register-blocked: one wave computes a 64x64 tile (4x4 grid of
// 16x16 WMMA accumulators) -> 16 WMMAs per 16 loads per K-step.
// grid=(2,64), blk=256 (8 waves); 16 waves cover the 256x256 output.
// ---------------------------------------------------------------------------
__global__ void __launch_bounds__(256) gram_kernel(const float* __restrict__ X,
                                                   float* __restrict__ A,
                                                   float* __restrict__ Z) {
    const int b    = blockIdx.y;
    const int w    = blockIdx.x * 8 + (threadIdx.x >> 5);   // 0..15
    const int lane = threadIdx.x & 31;
    const int ll   = lane & 15;
    const int hi   = lane >> 4;
    const int i0   = (w >> 2) * 64;
    const int j0   = (w & 3) * 64;

    const float* Xb = X + (size_t)b * M * N;
    v8f acc[4][4] = {};
    for (int k0 = 0; k0 < M; k0 += 4) {
        const float* r0 = Xb + (size_t)(k0 + 2 * hi) * N;   // row k0+2*hi of X
        __builtin_prefetch(r0 + 16 * N, 0, 1);              // global_prefetch
        v2f a[4], bb[4];
#pragma unroll
        for (int s = 0; s < 4; ++s) {
            a[s].x  = r0[i0 + 16 * s + ll];                 // (X^T)[i, k]
            a[s].y  = r0[N + i0 + 16 * s + ll];
            bb[s].x = r0[j0 + 16 * s + ll];                 // X[k, j]
            bb[s].y = r0[N + j0 + 16 * s + ll];
        }
#pragma unroll
        for (int si = 0; si < 4; ++si)
#pragma unroll
            for (int sj = 0; sj < 4; ++sj)
                acc[si][sj] = wmma_f32_4(a[si], bb[sj], acc[si][sj]);
    }
    float* Ab = A + (size_t)b * N * N;
    float* Zb = Z + (size_t)b * N * N;
#pragma unroll
    for (int si = 0; si < 4; ++si)
#pragma unroll
        for (int sj = 0; sj < 4; ++sj)
#pragma unroll
            for (int r = 0; r < 8; ++r) {
                const int row = i0 + 16 * si + r + 8 * hi;
                const int col = j0 + 16 * sj + ll;
                Ab[row * N + col] = acc[si][sj][r];
                Zb[row * N + col] = acc[si][sj][r];
            }
}

// ---------------------------------------------------------------------------
// Kernel 2: Z = A * Q per batch (256^3 GEMM), 64x64 tile per wave.
// A-side fragment pair is contiguous -> single b64 load per lane.
// grid=(2,64), blk=256.
// ---------------------------------------------------------------------------
__global__ void __launch_bounds__(256) gemm_AQ_kernel(const float* __restrict__ A,
                                                      const float* __restrict__ Q,
                                                      float* __restrict__ Z) {
    const int b    = blockIdx.y;
    const int w    = blockIdx.x * 8 + (threadIdx.x >> 5);   // 0..15
    const int lane = threadIdx.x & 31;
    const int ll   = lane & 15;
    const int hi   = lane >> 4;
    const int i0   = (w >> 2) * 64;
    const int j0   = (w & 3) * 64;

    const float* Ab = A + (size_t)b * N * N;
    const float* Qb = Q + (size_t)b * N * N;
    v8f acc[4][4] = {};
    for (int k0 = 0; k0 < N; k0 += 4) {
        v2f a[4], bb[4];
        const float* br = Qb + (size_t)(k0 + 2 * hi) * N;
#pragma unroll
        for (int s = 0; s < 4; ++s) {
            a[s] = *(const v2f*)(Ab + (size_t)(i0 + 16 * s + ll) * N +
                                 (k0 + 2 * hi));            // b64 load
            bb[s].x = br[j0 + 16 * s + ll];
            bb[s].y = br[N + j0 + 16 * s + ll];
        }
#pragma unroll
        for (int si = 0; si < 4; ++si)
#pragma unroll
            for (int sj = 0; sj < 4; ++sj)
                acc[si][sj] = wmma_f32_4(a[si], bb[sj], acc[si][sj]);
    }
    float* Zb = Z + (size_t)b * N * N;
#pragma unroll
    for (int si = 0; si < 4; ++si)
#pragma unroll
        for (int sj = 0; sj < 4; ++sj)
#pragma unroll
            for (int r = 0; r < 8; ++r)
                Zb[(i0 + 16 * si + r + 8 * hi) * N + (j0 + 16 * sj + ll)] =
                    acc[si][sj][r];
}

// ---------------------------------------------------------------------------
// Kernel 3: Modified Gram-Schmidt QR of Z (256x256) per batch.
// One workgroup (8 wave32) per batch; Z staged column-major in LDS with a
// 257-float padded column stride (bank-conflict-free column access).
// Writes Q (orthonormal) and diag(R) (column norms, >= 0).
// Dynamic LDS: (257*256 + 32) floats ~= 263 KB of the 320 KB WGP LDS.
// ---------------------------------------------------------------------------
#define ZSTRIDE 257
__global__ void __launch_bounds__(256) mgs_qr_kernel(const float* __restrict__ Z,
                                                     float* __restrict__ Q,
                                                     float* __restrict__ rdiag) {
    extern __shared__ float smem[];
    float* Zt  = smem;                 // [256 cols][257] column-major, padded
    float* red = smem + ZSTRIDE * N;   // [32] cross-wave reduction scratch

    const int b    = blockIdx.x;
    const int t    = threadIdx.x;      // 0..255
    const int lane = t & 31;
    const int wid  = t >> 5;           // 0..7

    const float* Zb = Z + (size_t)b * N * N;
    // Load + transpose: thread t owns column t of Zt; global reads coalesced.
    for (int r = 0; r < N; ++r)
        Zt[t * ZSTRIDE + r] = Zb[r * N + t];
    __syncthreads();

    for (int k = 0; k < N; ++k) {
        // ---- column-k norm (block reduction over 256 elements) ----
        float v = Zt[k * ZSTRIDE + t];
        float s = v * v;
#pragma unroll
        for (int off = 16; off; off >>= 1)
            s += __shfl_xor(s, off, 32);
        if (lane == 0) red[wid] = s;
        __syncthreads();
        const float tot  = red[0] + red[1] + red[2] + red[3] +
                           red[4] + red[5] + red[6] + red[7];
        const float norm = sqrtf(tot);
        const float inv  = (norm > 0.0f) ? (1.0f / norm) : 0.0f;
        Zt[k * ZSTRIDE + t] = v * inv;                 // q_k in place
        if (t == 0) rdiag[b * N + k] = norm;           // diag(R), |.| implicit
        __syncthreads();

        // ---- rank-1 update of trailing columns: one wave per column j ----
        for (int j = k + 1 + wid; j < N; j += 8) {
            float dot = 0.0f;
#pragma unroll
            for (int e = 0; e < 8; ++e)
                dot += Zt[k * ZSTRIDE + lane + 32 * e] *
                       Zt[j * ZSTRIDE + lane + 32 * e];
#pragma unroll
            for (int off = 16; off; off >>= 1)
                dot += __shfl_xor(dot, off, 32);
#pragma unroll
            for (int e = 0; e < 8; ++e)
                Zt[j * ZSTRIDE + lane + 32 * e] -=
                    dot * Zt[k * ZSTRIDE + lane + 32 * e];
        }
        __syncthreads();
    }

    // Write Q back row-major (coalesced: thread t = column index).
    float* Qb = Q + (size_t)b * N * N;
    for (int r = 0; r < N; ++r)
        Qb[r * N + t] = Zt[t * ZSTRIDE + r];
}

// ---------------------------------------------------------------------------
// Kernel 4: U = X * V, fused scale by 1/(sqrt(diagR)+1e-10).
// 64x64 tile per wave; grid=(8,64), blk=256.
// ---------------------------------------------------------------------------
__global__ void __launch_bounds__(256) xu_kernel(const float* __restrict__ X,
                                                 const float* __restrict__ V,
                                                 const float* __restrict__ rdiag,
                                                 float* __restrict__ U) {
    const int b    = blockIdx.y;
    const int w    = blockIdx.x * 8 + (threadIdx.x >> 5);   // 0..63
    const int lane = threadIdx.x & 31;
    const int ll   = lane & 15;
    const int hi   = lane >> 4;
    const int i0   = (w >> 2) * 64;    // 16 M-tiles of 64 -> 1024 rows
    const int j0   = (w & 3) * 64;     // 4 N-tiles of 64 -> 256 cols

    const float* Xb = X + (size_t)b * M * N;
    const float* Vb = V + (size_t)b * N * N;
    v8f acc[4][4] = {};
    for (int k0 = 0; k0 < N; k0 += 4) {
        v2f a[4], bb[4];
        const float* br = Vb + (size_t)(k0 + 2 * hi) * N;
#pragma unroll
        for (int s = 0; s < 4; ++s) {
            a[s] = *(const v2f*)(Xb + (size_t)(i0 + 16 * s + ll) * N +
                                 (k0 + 2 * hi));            // b64 load
            bb[s].x = br[j0 + 16 * s + ll];
            bb[s].y = br[N + j0 + 16 * s + ll];
        }
#pragma unroll
        for (int si = 0; si < 4; ++si)
#pragma unroll
            for (int sj = 0; sj < 4; ++sj)
                acc[si][sj] = wmma_f32_4(a[si], bb[sj], acc[si][sj]);
    }
    float* Ub = U + (size_t)b * M * N;
#pragma unroll
    for (int sj = 0; sj < 4; ++sj) {
        const int   col = j0 + 16 * sj + ll;
        const float scl = 1.0f / (sqrtf(rdiag[b * N + col]) + 1e-10f);
#pragma unroll
        for (int si = 0; si < 4; ++si)
#pragma unroll
            for (int r = 0; r < 8; ++r)
                Ub[(size_t)(i0 + 16 * si + r + 8 * hi) * N + col] =
                    acc[si][sj][r] * scl;
    }
}

// ---------------------------------------------------------------------------
// Kernel 5: S = sqrt(diagR)
// ---------------------------------------------------------------------------
__global__ void s_kernel(const float* __restrict__ rdiag, float* __restrict__ S) {
    const int i = blockIdx.x * blockDim.x + threadIdx.x;
    if (i < BATCH * N) S[i] = sqrtf(rdiag[i]);
}

// ---------------------------------------------------------------------------
extern "C" void kernel_launch(void* const* d_in, const int* in_sizes, int n_in,
                              void* d_out, int out_size, void* d_ws, size_t ws_size,
                              hipStream_t stream) {
    const float* X = (const float*)d_in[0];

    // d_out layout: U [64*1024*256] | S [64*256] | V [64*256*256]
    float* U = (float*)d_out;
    float* S = U + (size_t)BATCH * M * N;
    float* V = S + (size_t)BATCH * N;          // Q iterates live here

    // d_ws layout: A | Z | rdiag  (~34 MB)
    float* A  = (float*)d_ws;
    float* Z  = A + (size_t)BATCH * N * N;
    float* rd = Z + (size_t)BATCH * N * N;

    const dim3 blk(256);
    const size_t qr_lds = (size_t)(ZSTRIDE * N + 32) * sizeof(float);

    // A = X^T X ; Z = A (folds iteration 0's  Z = A @ I)
    gram_kernel<<<dim3(2, BATCH), blk, 0, stream>>>(X, A, Z);
    // Iteration 1 QR
    mgs_qr_kernel<<<dim3(BATCH), blk, qr_lds, stream>>>(Z, V, rd);
    // Iterations 2..8: Z = A*Q ; Q,R = qr(Z)
    for (int it = 1; it < MAX_ITER; ++it) {
        gemm_AQ_kernel<<<dim3(2, BATCH), blk, 0, stream>>>(A, V, Z);
        mgs_qr_kernel<<<dim3(BATCH), blk, qr_lds, stream>>>(Z, V, rd);
    }
    // S = sqrt(diag R) ; U = X V / (S + eps)
    s_kernel<<<dim3(BATCH), dim3(256), 0, stream>>>(rd, S);
    xu_kernel<<<dim3(8, BATCH), blk, 0, stream>>>(X, V, rd, U);
}